// SimpleLink5G_85598698209311
// MI455X (gfx1250) — compile-verified
//
#include <hip/hip_runtime.h>
#include <hip/hip_bf16.h>

// ---------------------------------------------------------------------------
// Fused 5G link-sim kernel for MI455X (gfx1250, wave32):
//   LDPC encode -> 16QAM + AWGN -> WMMA-based APP demap -> 20x sum-product BP
// One workgroup per codeword; all BP state LDS-resident; c2v messages live in
// VGPRs (check-centric edge grouping makes the check reduction thread-local).
// Demap metric (|p|^2 - 2 y.p) computed with V_WMMA_F32_16X16X4_F32.
// ---------------------------------------------------------------------------

#define K_INFO 1056
#define N_CODE 2112
#define M_CHK  1056
#define DV     5
#define NBPS   4
#define NSYM   528            // N_CODE / NBPS
#define NITER  20
#define BATCH  1024
#define NPTS   16
#define TPB    352            // 11 waves of 32
#define WAVES  (TPB / 32)
#define CPT    3              // checks per thread: 352*3 = 1056
#define NTILE  (NSYM / 16)    // 33 WMMA tiles of 16 symbols

typedef float v2f __attribute__((ext_vector_type(2)));
typedef float v8f __attribute__((ext_vector_type(8)));

__global__ __launch_bounds__(TPB) void link5g_fused_kernel(
    const int*   __restrict__ bits,       // [B, K]
    const int*   __restrict__ a_idx,      // [M, DV]
    const float* __restrict__ points_re,  // [16]
    const float* __restrict__ points_im,  // [16]
    const float* __restrict__ noise_re,   // [B, NSYM]
    const float* __restrict__ noise_im,   // [B, NSYM]
    const float* __restrict__ ebno_db,    // [1]
    float*       __restrict__ out)        // [2, B, K]
{
    __shared__ int   s_cw[N_CODE];            // codeword bits
    __shared__ float s_llr[N_CODE];           // channel LLRs
    __shared__ float s_vn[N_CODE];            // VN marginal accumulator
    __shared__ float s_metric[NSYM * NPTS];   // demap metrics lg' = -(|p|^2-2y.p)/no
    __shared__ float s_pre[NPTS], s_pim[NPTS], s_psq[NPTS];

    const int b    = blockIdx.x;
    const int tid  = threadIdx.x;
    const int lane = tid & 31;
    const int wave = tid >> 5;

    // N0 = 1 / (10^(EbN0/10) * coderate * NBPS), coderate = 0.5
    const float eb     = ebno_db[0];
    const float no     = 1.0f / (__expf(eb * 0.2302585093f) * 0.5f * (float)NBPS);
    const float inv_no = 1.0f / no;
    const float sigma  = __fsqrt_rn(no * 0.5f);

    if (tid < NPTS) {
        float pr = points_re[tid], pi = points_im[tid];
        s_pre[tid] = pr; s_pim[tid] = pi; s_psq[tid] = pr * pr + pi * pi;
    }
    // info bits -> LDS, and emit first output half (bits as f32)
    for (int k = tid; k < K_INFO; k += TPB) {
        int bit = bits[(size_t)b * K_INFO + k];
        s_cw[k] = bit;
        out[(size_t)b * K_INFO + k] = (float)bit;
    }
    __syncthreads();

    // ---- per-thread check-local graph: 3 checks x 6 edges, all in VGPRs ----
    int   vn[CPT][DV + 1];
    float c2v[CPT][DV + 1];
    #pragma unroll
    for (int c = 0; c < CPT; ++c) {
        const int ck = tid + c * TPB;
        #pragma unroll
        for (int j = 0; j < DV; ++j) vn[c][j] = a_idx[ck * DV + j];
        vn[c][DV] = K_INFO + ck;                 // identity-part parity edge
        #pragma unroll
        for (int j = 0; j <= DV; ++j) c2v[c][j] = 0.0f;
    }

    // ---- LDPC encode: parity_ck = XOR of the DV info bits of row ck ----
    #pragma unroll
    for (int c = 0; c < CPT; ++c) {
        const int ck = tid + c * TPB;
        int p = 0;
        #pragma unroll
        for (int j = 0; j < DV; ++j) p ^= s_cw[vn[c][j]];
        s_cw[K_INFO + ck] = p;
    }
    __syncthreads();

    // ---- QAM map + AWGN + WMMA demap metric -------------------------------
    // D[m][p] = sum_k A[m][k]*B[k][p] = |p|^2 - 2*(y_re*p_re + y_im*p_im)
    //   A row m:  [-2*y_re, -2*y_im, 1, 0]   (16x4 f32)
    //   B col p:  [p_re, p_im, |p|^2, 0]^T   (4x16 f32)
    // f32 A 16x4 layout: lanes0-15 {K0,K1}, lanes16-31 {K2,K3} (VGPR0,VGPR1).
    // B assumed mirrored: lanes0-15 rows {K0,K1}, lanes16-31 rows {K2,K3}.
    // All 11 waves run exactly 3 tiles, branch-free -> EXEC all-1s at WMMA.
    for (int tile = wave; tile < NTILE; tile += WAVES) {
        const int  m  = lane & 15;
        const int  s  = tile * 16 + m;
        const bool lo = (lane < 16);

        const int sym = (s_cw[s * 4 + 0] << 3) | (s_cw[s * 4 + 1] << 2) |
                        (s_cw[s * 4 + 2] << 1) |  s_cw[s * 4 + 3];
        const float yre = s_pre[sym] + sigma * noise_re[(size_t)b * NSYM + s];
        const float yim = s_pim[sym] + sigma * noise_im[(size_t)b * NSYM + s];

        v2f a, bb;
        a.x  = lo ? (-2.0f * yre) : 1.0f;        // K0 | K2
        a.y  = lo ? (-2.0f * yim) : 0.0f;        // K1 | K3
        bb.x = lo ? s_pre[m] : s_psq[m];         // K0 row | K2 row, col m
        bb.y = lo ? s_pim[m] : 0.0f;             // K1 row | K3 row, col m

        v8f cacc = {};
        v8f d = __builtin_amdgcn_wmma_f32_16x16x4_f32(
            /*neg_a=*/false, a, /*neg_b=*/false, bb,
            /*c_mod=*/(short)0, cacc, /*reuse_a=*/false, /*reuse_b=*/false);

        // D layout: VGPR r -> row (r + 8*(lane>=16)), col = lane&15
        const int half8 = (lane >> 4) << 3;
        #pragma unroll
        for (int r = 0; r < 8; ++r) {
            const int mr = r + half8;
            s_metric[(tile * 16 + mr) * NPTS + m] = -d[r] * inv_no;
        }
    }
    __syncthreads();

    // ---- exact logsumexp APP demap (|y|^2 term cancels in l0-l1) ----------
    for (int s = tid; s < NSYM; s += TPB) {
        float mg[NPTS];
        #pragma unroll
        for (int p = 0; p < NPTS; ++p) mg[p] = s_metric[s * NPTS + p];
        #pragma unroll
        for (int j = 0; j < NBPS; ++j) {
            float m0 = -3.0e38f, m1 = -3.0e38f;
            #pragma unroll
            for (int p = 0; p < NPTS; ++p) {
                if ((p >> (3 - j)) & 1) m1 = fmaxf(m1, mg[p]);
                else                    m0 = fmaxf(m0, mg[p]);
            }
            float e0 = 0.0f, e1 = 0.0f;
            #pragma unroll
            for (int p = 0; p < NPTS; ++p) {
                if ((p >> (3 - j)) & 1) e1 += __expf(mg[p] - m1);
                else                    e0 += __expf(mg[p] - m0);
            }
            s_llr[s * NBPS + j] = (m0 + __logf(e0)) - (m1 + __logf(e1));
        }
    }
    __syncthreads();

    // ---- flooding sum-product BP, c2v in registers, marginals via LDS -----
    for (int it = 0; it < NITER; ++it) {
        for (int n = tid; n < N_CODE; n += TPB) s_vn[n] = s_llr[n];
        __syncthreads();
        #pragma unroll
        for (int c = 0; c < CPT; ++c)
            #pragma unroll
            for (int j = 0; j <= DV; ++j)
                atomicAdd(&s_vn[vn[c][j]], c2v[c][j]);   // ds_add_f32
        __syncthreads();
        #pragma unroll
        for (int c = 0; c < CPT; ++c) {
            float lt[DV + 1]; int sg[DV + 1];
            float cs = 0.0f; int ns = 0;
            #pragma unroll
            for (int j = 0; j <= DV; ++j) {
                const float v2c = s_vn[vn[c][j]] - c2v[c][j];
                const float av  = fminf(fmaxf(fabsf(v2c), 1e-6f), 20.0f);
                const float t   = __expf(-av);
                lt[j] = __logf((1.0f - t) / (1.0f + t));   // log tanh(av/2) <= 0
                sg[j] = (v2c < 0.0f) ? 1 : 0;
                cs += lt[j]; ns += sg[j];
            }
            #pragma unroll
            for (int j = 0; j <= DV; ++j) {
                const float ex = __expf(fminf(cs - lt[j], -1e-7f));
                const float sp = 1.0f - 2.0f * (float)((ns - sg[j]) & 1);
                // 2*atanh(ex) = log((1+ex)/(1-ex))
                c2v[c][j] = sp * __logf((1.0f + ex) / (1.0f - ex));
            }
        }
        __syncthreads();
    }

    // ---- final marginal + hard decision (second output half) --------------
    for (int n = tid; n < N_CODE; n += TPB) s_vn[n] = s_llr[n];
    __syncthreads();
    #pragma unroll
    for (int c = 0; c < CPT; ++c)
        #pragma unroll
        for (int j = 0; j <= DV; ++j)
            atomicAdd(&s_vn[vn[c][j]], c2v[c][j]);
    __syncthreads();
    for (int k = tid; k < K_INFO; k += TPB) {
        out[(size_t)BATCH * K_INFO + (size_t)b * K_INFO + k] =
            (s_vn[k] < 0.0f) ? 1.0f : 0.0f;
    }
}

extern "C" void kernel_launch(void* const* d_in, const int* in_sizes, int n_in,
                              void* d_out, int out_size, void* d_ws, size_t ws_size,
                              hipStream_t stream) {
    (void)in_sizes; (void)n_in; (void)out_size; (void)d_ws; (void)ws_size;
    const int*   bits      = (const int*)  d_in[0];
    const int*   a_idx     = (const int*)  d_in[1];
    // d_in[2] = edge_cn, d_in[3] = edge_vn: unused (structure derived from a_idx)
    const float* points_re = (const float*)d_in[4];
    const float* points_im = (const float*)d_in[5];
    const float* noise_re  = (const float*)d_in[6];
    const float* noise_im  = (const float*)d_in[7];
    const float* ebno      = (const float*)d_in[8];
    float* out = (float*)d_out;

    link5g_fused_kernel<<<BATCH, TPB, 0, stream>>>(
        bits, a_idx, points_re, points_im, noise_re, noise_im, ebno, out);
}